// GCNLayer_49503793054215
// MI455X (gfx1250) — compile-verified
//
#include <hip/hip_runtime.h>
#include <hip/hip_bf16.h>

typedef __attribute__((ext_vector_type(2))) float v2f;
typedef __attribute__((ext_vector_type(8))) float v8f;

#define N_NODES 100000
#define N_EDGES 1600000
#define CH 64
#define NUM_GRAPHS 64

// ---------------------------------------------------------------------------
// 1) init: deg = 1.0 (self loop), pool sums / counts = 0
// ---------------------------------------------------------------------------
__global__ void gcn_init_kernel(float* __restrict__ deg,
                                float* __restrict__ poolsum,
                                float* __restrict__ poolcnt) {
    int i = blockIdx.x * blockDim.x + threadIdx.x;
    if (i < N_NODES) deg[i] = 1.0f;
    if (i < NUM_GRAPHS * CH) poolsum[i] = 0.0f;
    if (i < NUM_GRAPHS) poolcnt[i] = 0.0f;
}

// ---------------------------------------------------------------------------
// 2) degree accumulation over destination nodes
// ---------------------------------------------------------------------------
__global__ void gcn_degree_kernel(const int* __restrict__ edge_index,
                                  float* __restrict__ deg) {
    int e = blockIdx.x * blockDim.x + threadIdx.x;
    if (e >= N_EDGES) return;
    int d = edge_index[N_EDGES + e];  // edge_index[1][e]
    atomicAdd(&deg[d], 1.0f);
}

// ---------------------------------------------------------------------------
// 3) h = x @ W via V_WMMA_F32_16X16X4_F32 (exact f32 matrix path)
//    Block = 128 threads = 4 waves. Each wave: one 16-row tile x all 64 cols.
//    W is staged in LDS pre-interleaved by K-pairs:
//      lW[k2][n] = { W[2*k2][n], W[2*k2+1][n] }   (8-byte elements)
//    so every B fragment is ONE ds_load_b64 into an even-aligned VGPR pair —
//    no register shuffling between DS returns and the WMMA operands.
//    Rows are clamped (not masked) so EXEC stays all-ones, as WMMA requires.
// ---------------------------------------------------------------------------
__global__ void gcn_gemm_wmma_kernel(const float* __restrict__ x,
                                     const float* __restrict__ W,
                                     float* __restrict__ h) {
    __shared__ v2f lW[(CH / 2) * CH];  // 32 x 64 x 8B = 16 KB of 320 KB/WGP LDS

    for (int i = threadIdx.x; i < (CH / 2) * CH; i += 128) {
        int k2 = i >> 6;       // K pair index 0..31
        int n  = i & 63;       // column
        v2f w;
        w.x = W[(2 * k2) * CH + n];
        w.y = W[(2 * k2 + 1) * CH + n];
        lW[i] = w;
    }
    __syncthreads();

    const int wave = threadIdx.x >> 5;
    const int lane = threadIdx.x & 31;
    const int m    = lane & 15;        // row-within-tile / col-within-tile
    const int kh   = (lane >> 4) * 2;  // upper half-wave handles K pair {2,3}

    const int row0 = (blockIdx.x * 4 + wave) * 16;
    if (row0 >= N_NODES) return;  // whole-wave early out (uniform branch)

    int row = row0 + m;
    if (row >= N_NODES) row = N_NODES - 1;  // clamp: keep EXEC all-ones

    v8f acc0 = {}, acc1 = {}, acc2 = {}, acc3 = {};

#pragma unroll
    for (int k4 = 0; k4 < CH; k4 += 4) {
        // A fragment: x[row][k4+kh .. k4+kh+1]  -> one global_load_b64
        v2f a = *(const v2f*)(x + row * CH + k4 + kh);

        // B fragments: interleaved pair row (k4+kh)/2, column tile*16 + m
        const int kp = ((k4 + kh) >> 1) * CH + m;  // k4+kh is always even
        v2f b0 = lW[kp + 0];
        v2f b1 = lW[kp + 16];
        v2f b2 = lW[kp + 32];
        v2f b3 = lW[kp + 48];

        acc0 = __builtin_amdgcn_wmma_f32_16x16x4_f32(false, a, false, b0, (short)0, acc0, false, false);
        acc1 = __builtin_amdgcn_wmma_f32_16x16x4_f32(false, a, false, b1, (short)0, acc1, false, false);
        acc2 = __builtin_amdgcn_wmma_f32_16x16x4_f32(false, a, false, b2, (short)0, acc2, false, false);
        acc3 = __builtin_amdgcn_wmma_f32_16x16x4_f32(false, a, false, b3, (short)0, acc3, false, false);
    }

    // C/D layout: VGPR i -> M = i + (lane>=16 ? 8 : 0), N = lane&15
    const int mbase = row0 + ((lane >> 4) << 3);
#pragma unroll
    for (int i = 0; i < 8; ++i) {
        const int r = mbase + i;
        if (r < N_NODES) {
            h[r * CH + 0  + m] = acc0[i];
            h[r * CH + 16 + m] = acc1[i];
            h[r * CH + 32 + m] = acc2[i];
            h[r * CH + 48 + m] = acc3[i];
        }
    }
}

// ---------------------------------------------------------------------------
// 4) dinv = rsqrt(deg); agg (in d_out) initialized with self-loop term
//    agg[i][c] = dinv[i]^2 * h[i][c]
// ---------------------------------------------------------------------------
__global__ void gcn_selfloop_kernel(const float* __restrict__ deg,
                                    const float* __restrict__ h,
                                    float* __restrict__ dinv,
                                    float* __restrict__ agg) {
    long long idx = (long long)blockIdx.x * blockDim.x + threadIdx.x;
    long long node = idx >> 6;
    int c = (int)(idx & 63);
    if (node >= N_NODES) return;
    float di = rsqrtf(deg[node]);  // deg >= 1 always (self loop)
    if (c == 0) dinv[node] = di;
    agg[node * CH + c] = di * di * h[node * CH + c];
}

// ---------------------------------------------------------------------------
// 5) edge scatter: 64 threads per edge (one channel each).
//    Gathers of h are L2-resident (25.6 MB << 192 MB L2); this stage is the
//    roofline limiter (~870 MB of gather + atomic-RMW traffic).
// ---------------------------------------------------------------------------
__global__ void gcn_edge_kernel(const int* __restrict__ edge_index,
                                const float* __restrict__ h,
                                const float* __restrict__ dinv,
                                float* __restrict__ agg) {
    long long idx = (long long)blockIdx.x * blockDim.x + threadIdx.x;
    long long e = idx >> 6;
    int c = (int)(idx & 63);
    if (e >= N_EDGES) return;
    int s = edge_index[e];             // edge_index[0][e]  (src)
    int d = edge_index[N_EDGES + e];   // edge_index[1][e]  (dst)
    float nrm = dinv[s] * dinv[d];
    atomicAdd(&agg[(long long)d * CH + c], nrm * h[(long long)s * CH + c]);
}

// ---------------------------------------------------------------------------
// 6) out = relu(agg + b) in place; accumulate pool sums & counts
// ---------------------------------------------------------------------------
__global__ void gcn_relu_pool_kernel(const float* __restrict__ b,
                                     const int* __restrict__ batch,
                                     float* __restrict__ out,
                                     float* __restrict__ poolsum,
                                     float* __restrict__ poolcnt) {
    long long idx = (long long)blockIdx.x * blockDim.x + threadIdx.x;
    long long node = idx >> 6;
    int c = (int)(idx & 63);
    if (node >= N_NODES) return;
    float v = out[node * CH + c] + b[c];
    v = fmaxf(v, 0.0f);
    out[node * CH + c] = v;
    int g = batch[node];
    atomicAdd(&poolsum[g * CH + c], v);
    if (c == 0) atomicAdd(&poolcnt[g], 1.0f);
}

// ---------------------------------------------------------------------------
// 7) pooled = sums / max(counts, 1)
// ---------------------------------------------------------------------------
__global__ void gcn_pool_final_kernel(const float* __restrict__ poolsum,
                                      const float* __restrict__ poolcnt,
                                      float* __restrict__ pooled) {
    int i = blockIdx.x * blockDim.x + threadIdx.x;
    if (i >= NUM_GRAPHS * CH) return;
    float cnt = poolcnt[i >> 6];
    pooled[i] = poolsum[i] / fmaxf(cnt, 1.0f);
}

// ---------------------------------------------------------------------------
extern "C" void kernel_launch(void* const* d_in, const int* in_sizes, int n_in,
                              void* d_out, int out_size, void* d_ws, size_t ws_size,
                              hipStream_t stream) {
    const float* x          = (const float*)d_in[0];
    const int*   edge_index = (const int*)d_in[1];
    const int*   batch      = (const int*)d_in[2];
    const float* W          = (const float*)d_in[3];
    const float* b          = (const float*)d_in[4];

    float* out    = (float*)d_out;                       // [N, 64]
    float* pooled = out + (long long)N_NODES * CH;       // [64, 64]

    // workspace layout (floats)
    float* ws      = (float*)d_ws;
    float* h       = ws;                                  // N*64
    float* deg     = h + (long long)N_NODES * CH;         // N
    float* dinv    = deg + N_NODES;                       // N
    float* poolsum = dinv + N_NODES;                      // 64*64
    float* poolcnt = poolsum + NUM_GRAPHS * CH;           // 64

    // 1) init
    gcn_init_kernel<<<(N_NODES + 255) / 256, 256, 0, stream>>>(deg, poolsum, poolcnt);
    // 2) degrees
    gcn_degree_kernel<<<(N_EDGES + 255) / 256, 256, 0, stream>>>(edge_index, deg);
    // 3) h = x @ W (WMMA f32)
    {
        int nblocks = (N_NODES + 63) / 64;  // 4 waves/block * 16 rows/wave
        gcn_gemm_wmma_kernel<<<nblocks, 128, 0, stream>>>(x, W, h);
    }
    // 4) dinv + self-loop init of agg (= out buffer)
    {
        long long t = (long long)N_NODES * CH;
        gcn_selfloop_kernel<<<(unsigned)((t + 255) / 256), 256, 0, stream>>>(deg, h, dinv, out);
    }
    // 5) edge scatter
    {
        long long t = (long long)N_EDGES * CH;
        gcn_edge_kernel<<<(unsigned)((t + 255) / 256), 256, 0, stream>>>(edge_index, h, dinv, out);
    }
    // 6) relu + bias + pooling accumulation
    {
        long long t = (long long)N_NODES * CH;
        gcn_relu_pool_kernel<<<(unsigned)((t + 255) / 256), 256, 0, stream>>>(b, batch, out, poolsum, poolcnt);
    }
    // 7) pool finalize
    gcn_pool_final_kernel<<<(NUM_GRAPHS * CH + 255) / 256, 256, 0, stream>>>(poolsum, poolcnt, pooled);
}